// new_sim_88210038326190
// MI455X (gfx1250) — compile-verified
//
#include <hip/hip_runtime.h>

// Elementwise:  j = 1 - log10(1024*|x-y| + 1)
// (reference: sqrt(1024^2 * d^2) == 1024*|d| exactly)
//
// Memory-bound streaming kernel: 12 B/elem, ~402 MB total -> ~17us floor at
// 23.3 TB/s. No matmul structure -> WMMA/TDM inapplicable; optimal CDNA5 path
// is b128 vector memory ops with non-temporal hints + v_log_f32.

typedef float v4f __attribute__((ext_vector_type(4)));

#define LOG10_OF_2 0.30102999566398119521f

__device__ __forceinline__ float jfun(float xv, float yv) {
  float d = __builtin_fabsf(xv - yv);              // v_sub + abs modifier
  float v = __builtin_fmaf(1024.0f, d, 1.0f);      // v_fma_f32
  // 1 - log10(v) = 1 - log10(2) * log2(v)
  return __builtin_fmaf(-LOG10_OF_2, __builtin_amdgcn_logf(v), 1.0f); // v_log_f32 + v_fma
}

__device__ __forceinline__ v4f jfun4(v4f a, v4f b) {
  v4f r;
  r.x = jfun(a.x, b.x);
  r.y = jfun(a.y, b.y);
  r.z = jfun(a.z, b.z);
  r.w = jfun(a.w, b.w);
  return r;
}

#define TPB 256
#define V4_PER_THREAD 4   // 4 independent b128 loads -> clause-friendly, 64B/thread/array

__global__ __launch_bounds__(TPB) void jindex_stream_kernel(
    const float* __restrict__ x, const float* __restrict__ y,
    float* __restrict__ out, long long n) {
  const long long n4 = n >> 2;
  const v4f* __restrict__ x4 = (const v4f*)x;
  const v4f* __restrict__ y4 = (const v4f*)y;
  v4f* __restrict__ o4 = (v4f*)out;

  const long long span = (long long)TPB * V4_PER_THREAD;
  const long long base = (long long)blockIdx.x * span + threadIdx.x;

  if (base + (long long)(V4_PER_THREAD - 1) * TPB < n4) {
    // Steady state: unguarded -> compiler can clause 8 global_load_b128 (NT)
    v4f a0 = __builtin_nontemporal_load(x4 + base + 0 * TPB);
    v4f b0 = __builtin_nontemporal_load(y4 + base + 0 * TPB);
    v4f a1 = __builtin_nontemporal_load(x4 + base + 1 * TPB);
    v4f b1 = __builtin_nontemporal_load(y4 + base + 1 * TPB);
    v4f a2 = __builtin_nontemporal_load(x4 + base + 2 * TPB);
    v4f b2 = __builtin_nontemporal_load(y4 + base + 2 * TPB);
    v4f a3 = __builtin_nontemporal_load(x4 + base + 3 * TPB);
    v4f b3 = __builtin_nontemporal_load(y4 + base + 3 * TPB);
    __builtin_nontemporal_store(jfun4(a0, b0), o4 + base + 0 * TPB);
    __builtin_nontemporal_store(jfun4(a1, b1), o4 + base + 1 * TPB);
    __builtin_nontemporal_store(jfun4(a2, b2), o4 + base + 2 * TPB);
    __builtin_nontemporal_store(jfun4(a3, b3), o4 + base + 3 * TPB);
  } else {
    // Tail block: guarded per-vector
    #pragma unroll
    for (int k = 0; k < V4_PER_THREAD; ++k) {
      long long i = base + (long long)k * TPB;
      if (i < n4) {
        v4f a = __builtin_nontemporal_load(x4 + i);
        v4f b = __builtin_nontemporal_load(y4 + i);
        __builtin_nontemporal_store(jfun4(a, b), o4 + i);
      }
    }
    // Scalar remainder (n % 4) — zero for this shape but kept for safety
    if (blockIdx.x == 0 && (long long)threadIdx.x < (n & 3LL)) {
      long long i = (n4 << 2) + threadIdx.x;
      out[i] = jfun(x[i], y[i]);
    }
  }
}

extern "C" void kernel_launch(void* const* d_in, const int* in_sizes, int n_in,
                              void* d_out, int out_size, void* d_ws, size_t ws_size,
                              hipStream_t stream) {
  (void)n_in; (void)d_ws; (void)ws_size; (void)out_size;
  const float* x = (const float*)d_in[0];
  const float* y = (const float*)d_in[1];
  float* out = (float*)d_out;
  const long long n = (long long)in_sizes[0];     // 33,554,432

  const long long n4 = n >> 2;                    // float4 count
  const long long per_block = (long long)TPB * V4_PER_THREAD; // 1024 float4s
  const int blocks = (int)((n4 + per_block - 1) / per_block); // 8192
  jindex_stream_kernel<<<blocks, TPB, 0, stream>>>(x, y, out, n);
}